// CPGNodePairModel_48172353192362
// MI455X (gfx1250) — compile-verified
//
#include <hip/hip_runtime.h>
#include <hip/hip_bf16.h>

typedef __attribute__((ext_vector_type(2))) float v2f;
typedef __attribute__((ext_vector_type(8))) float v8f;

#define DH 256
#define BK 32        // K-chunk staged through LDS
#define MT 32        // rows per block (100000 % 32 == 0)
#define AP 36        // padded A-tile row stride (floats, %4==0)
#define WP 260       // padded W-tile row stride (floats, %4==0)

// ---------------------------------------------------------------------------
// Dense GEMM via V_WMMA_F32_16X16X4_F32:
//   C[M, 256] = (relu?)(A[M, K] @ W[K, 256] (+ bias))
// Block = 256 threads (8 waves), computes a 32-row x 256-col tile.
// Wave w owns cols [w*32, w*32+32): 2x2 grid of 16x16 WMMA accumulators.
// A (32xBK) and W (BKx256) chunks are staged in LDS; the inner loop is
// pure ds_load + v_wmma. Next chunk is prefetched into cache.
// ---------------------------------------------------------------------------
__global__ __launch_bounds__(256)
void gemm_wmma_f32(const float* __restrict__ A, const float* __restrict__ W,
                   const float* __restrict__ bias, float* __restrict__ C,
                   int K, int applyRelu) {
    __shared__ float As[MT][AP];       // 32 x 32 A tile (+pad)
    __shared__ float Ws[BK][WP];       // 32 x 256 W tile (+pad)

    const int tid   = threadIdx.x;
    const int wave  = tid >> 5;
    const int lane  = tid & 31;
    const int lanem = lane & 15;           // N index in tile / M index for A operand
    const int koff  = (lane >> 4) * 2;     // K split: lanes 0-15 -> K+0,1 ; 16-31 -> K+2,3
    const int row0  = blockIdx.x * MT;
    const int col0  = wave * 32;

    v8f acc00 = {}, acc01 = {}, acc10 = {}, acc11 = {};

    // cooperative-load index precompute
    const int arow = tid >> 3, acol4 = (tid & 7) * 4;          // A: 1 float4/thread
    const int wrow = tid >> 6, wcol4 = (tid & 63) * 4;         // W: 8 float4/thread

    for (int k0 = 0; k0 < K; k0 += BK) {
        // ---- stage A tile: 32x32 floats = 256 float4 (1 per thread) ----
        *(float4*)&As[arow][acol4] =
            *(const float4*)&A[(size_t)(row0 + arow) * K + k0 + acol4];
        // ---- stage W tile: 32x256 floats = 2048 float4 (8 per thread) ----
        #pragma unroll
        for (int j = 0; j < 8; ++j) {
            *(float4*)&Ws[wrow + j * 4][wcol4] =
                *(const float4*)&W[(size_t)(k0 + wrow + j * 4) * DH + wcol4];
        }
        // ---- prefetch next chunk into cache while we compute ----
        if (k0 + BK < K) {
            if (tid < MT)
                __builtin_prefetch(&A[(size_t)(row0 + tid) * K + k0 + BK], 0, 0);
            __builtin_prefetch(&W[(size_t)(k0 + BK) * DH + tid * 32], 0, 0);
        }
        __syncthreads();

        #pragma unroll
        for (int kk = 0; kk < BK; kk += 4) {
            // A operands (16x4 f32): lane L<16 holds (M=L, K=kk+0/1), L>=16 K=kk+2/3
            v2f a0, a1;
            a0.x = As[lanem][kk + koff];
            a0.y = As[lanem][kk + koff + 1];
            a1.x = As[16 + lanem][kk + koff];
            a1.y = As[16 + lanem][kk + koff + 1];
            // B operands (4x16 f32): row striped across lanes within a VGPR
            v2f b0, b1;
            b0.x = Ws[kk + koff][col0 + lanem];
            b0.y = Ws[kk + koff + 1][col0 + lanem];
            b1.x = Ws[kk + koff][col0 + 16 + lanem];
            b1.y = Ws[kk + koff + 1][col0 + 16 + lanem];

            acc00 = __builtin_amdgcn_wmma_f32_16x16x4_f32(
                false, a0, false, b0, (short)0, acc00, false, false);
            acc01 = __builtin_amdgcn_wmma_f32_16x16x4_f32(
                false, a0, false, b1, (short)0, acc01, false, false);
            acc10 = __builtin_amdgcn_wmma_f32_16x16x4_f32(
                false, a1, false, b0, (short)0, acc10, false, false);
            acc11 = __builtin_amdgcn_wmma_f32_16x16x4_f32(
                false, a1, false, b1, (short)0, acc11, false, false);
        }
        __syncthreads();
    }

    // C/D layout: VGPR r -> M = r (lanes 0-15), M = r+8 (lanes 16-31); N = lane%16
    const int rsub = (lane >> 4) * 8;
    float bs0 = 0.f, bs1 = 0.f;
    if (bias) { bs0 = bias[col0 + lanem]; bs1 = bias[col0 + 16 + lanem]; }
    #pragma unroll
    for (int r = 0; r < 8; ++r) {
        int rowA = row0 + rsub + r;          // row tile 0
        int rowB = rowA + 16;                // row tile 1
        float v00 = acc00[r] + bs0, v01 = acc01[r] + bs1;
        float v10 = acc10[r] + bs0, v11 = acc11[r] + bs1;
        if (applyRelu) {
            v00 = fmaxf(v00, 0.f); v01 = fmaxf(v01, 0.f);
            v10 = fmaxf(v10, 0.f); v11 = fmaxf(v11, 0.f);
        }
        C[(size_t)rowA * DH + col0 + lanem]      = v00;
        C[(size_t)rowA * DH + col0 + 16 + lanem] = v01;
        C[(size_t)rowB * DH + col0 + lanem]      = v10;
        C[(size_t)rowB * DH + col0 + 16 + lanem] = v11;
    }
}

// ---------------------------------------------------------------------------
// Degree / normalization: deg = 1 (self loop) + in-degree; dinv = rsqrt(deg)
// ---------------------------------------------------------------------------
__global__ void deg_init(float* __restrict__ deg, int n) {
    int i = blockIdx.x * blockDim.x + threadIdx.x;
    if (i < n) deg[i] = 1.0f;
}

__global__ void deg_edges(const int* __restrict__ dst, float* __restrict__ deg, int e) {
    int i = blockIdx.x * blockDim.x + threadIdx.x;
    if (i < e) atomicAdd(&deg[dst[i]], 1.0f);
}

__global__ void deg_finish(float* __restrict__ deg, int n) {
    int i = blockIdx.x * blockDim.x + threadIdx.x;
    if (i < n) deg[i] = rsqrtf(deg[i]);
}

// agg[i,:] = m[i,:] * dinv[i]^2   (self-loop term; initializes agg)
__global__ void agg_init(const float* __restrict__ m, const float* __restrict__ dinv,
                         float* __restrict__ agg, int n) {
    int i = blockIdx.x * blockDim.x + threadIdx.x;
    int node = i >> 6;
    if (node < n) {
        float s = dinv[node]; s *= s;
        float4 v = ((const float4*)m)[i];
        ((float4*)agg)[i] = make_float4(v.x * s, v.y * s, v.z * s, v.w * s);
    }
}

// Edge scatter: agg[dst,:] += m[src,:] * dinv[src]*dinv[dst]
// Block = 4 groups of 64 lanes; each lane moves one float4 per edge.
#define EPG 8
__global__ __launch_bounds__(256)
void edge_scatter(const float* __restrict__ m, float* __restrict__ agg,
                  const int* __restrict__ src, const int* __restrict__ dst,
                  const float* __restrict__ dinv, int e) {
    const int g = threadIdx.x >> 6;
    const int t = threadIdx.x & 63;
    int ei = blockIdx.x * (4 * EPG) + g;
    #pragma unroll
    for (int it = 0; it < EPG; ++it, ei += 4) {
        if (ei < e) {
            int s = src[ei], d = dst[ei];
            float nrm = dinv[s] * dinv[d];
            float4 v = ((const float4*)&m[(size_t)s * DH])[t];
            float* ap = &agg[(size_t)d * DH + t * 4];
            atomicAdd(ap + 0, v.x * nrm);
            atomicAdd(ap + 1, v.y * nrm);
            atomicAdd(ap + 2, v.z * nrm);
            atomicAdd(ap + 3, v.w * nrm);
        }
    }
}

// h[i,:] = relu(agg[i,:] + bg)
__global__ void layer_finish(const float* __restrict__ agg, const float* __restrict__ bg,
                             float* __restrict__ h, int n) {
    int i = blockIdx.x * blockDim.x + threadIdx.x;
    int node = i >> 6;
    if (node < n) {
        int c4 = (i & 63) * 4;
        float4 v = ((const float4*)agg)[i];
        const float4 b = *(const float4*)&bg[c4];
        ((float4*)h)[i] = make_float4(fmaxf(v.x + b.x, 0.f), fmaxf(v.y + b.y, 0.f),
                                      fmaxf(v.z + b.z, 0.f), fmaxf(v.w + b.w, 0.f));
    }
}

// ---------------------------------------------------------------------------
// Classifier: gather 16 node pairs, 512->256->128->2 MLP, single block.
// ---------------------------------------------------------------------------
__global__ __launch_bounds__(256)
void classifier(const float* __restrict__ h,
                const int* __restrict__ sids, const int* __restrict__ kids,
                const float* __restrict__ W1, const float* __restrict__ b1,
                const float* __restrict__ W2, const float* __restrict__ b2,
                const float* __restrict__ W3, const float* __restrict__ b3,
                float* __restrict__ out, int npg) {
    __shared__ float pair[16][512];
    __shared__ float z1[16][256];
    __shared__ float z2[16][128];
    const int tid = threadIdx.x;

    for (int i = tid; i < 16 * 512; i += 256) {
        int b = i >> 9, c = i & 511;
        int node = b * npg + ((c < 256) ? sids[b] : kids[b]);
        pair[b][c] = h[(size_t)node * DH + (c & 255)];
    }
    __syncthreads();

    for (int b = 0; b < 16; ++b) {
        float acc = b1[tid];
        for (int k = 0; k < 512; ++k) acc += pair[b][k] * W1[k * 256 + tid];
        z1[b][tid] = fmaxf(acc, 0.0f);
    }
    __syncthreads();

    if (tid < 128) {
        for (int b = 0; b < 16; ++b) {
            float acc = b2[tid];
            for (int k = 0; k < 256; ++k) acc += z1[b][k] * W2[k * 128 + tid];
            z2[b][tid] = fmaxf(acc, 0.0f);
        }
    }
    __syncthreads();

    if (tid < 32) {
        int b = tid >> 1, j = tid & 1;
        float acc = b3[j];
        for (int k = 0; k < 128; ++k) acc += z2[b][k] * W3[k * 2 + j];
        out[b * 2 + j] = acc;
    }
}

// ---------------------------------------------------------------------------
extern "C" void kernel_launch(void* const* d_in, const int* in_sizes, int n_in,
                              void* d_out, int out_size, void* d_ws, size_t ws_size,
                              hipStream_t stream) {
    const float* x    = (const float*)d_in[0];
    const int*   ei   = (const int*)d_in[1];   // [2, E] flat: src row then dst row
    // d_in[2] = batch (unused: graphs contiguous, equal-sized)
    const int*   sids = (const int*)d_in[3];
    const int*   kids = (const int*)d_in[4];
    const float* Wi   = (const float*)d_in[5];
    const float* bi   = (const float*)d_in[6];
    const float* Wg   = (const float*)d_in[7];
    const float* bg   = (const float*)d_in[8];
    const float* W1   = (const float*)d_in[9];
    const float* b1   = (const float*)d_in[10];
    const float* W2   = (const float*)d_in[11];
    const float* b2   = (const float*)d_in[12];
    const float* W3   = (const float*)d_in[13];
    const float* b3   = (const float*)d_in[14];
    float* out = (float*)d_out;

    const int DIN = 768;
    const int N = in_sizes[0] / DIN;      // 100000
    const int E = in_sizes[1] / 2;        // 1600000
    const int B = in_sizes[3];            // 16
    const int NPG = N / B;

    // workspace layout (floats): h | m | agg | dinv
    float* h    = (float*)d_ws;
    float* m    = h   + (size_t)N * DH;
    float* agg  = m   + (size_t)N * DH;
    float* dinv = agg + (size_t)N * DH;

    // GCN normalization
    deg_init  <<<(N + 255) / 256, 256, 0, stream>>>(dinv, N);
    deg_edges <<<(E + 255) / 256, 256, 0, stream>>>(ei + E, dinv, E);
    deg_finish<<<(N + 255) / 256, 256, 0, stream>>>(dinv, N);

    // input projection: h = relu(x @ Wi + bi)
    gemm_wmma_f32<<<N / MT, 256, 0, stream>>>(x, Wi, bi, h, DIN, 1);

    const int vec4 = N * (DH / 4);
    for (int l = 0; l < 3; ++l) {
        gemm_wmma_f32<<<N / MT, 256, 0, stream>>>(h, Wg + (size_t)l * DH * DH,
                                                  nullptr, m, DH, 0);
        agg_init<<<(vec4 + 255) / 256, 256, 0, stream>>>(m, dinv, agg, N);
        edge_scatter<<<(E + 4 * EPG - 1) / (4 * EPG), 256, 0, stream>>>(
            m, agg, ei, ei + E, dinv, E);
        layer_finish<<<(vec4 + 255) / 256, 256, 0, stream>>>(agg, bg + l * DH, h, N);
    }

    classifier<<<1, 256, 0, stream>>>(h, sids, kids, W1, b1, W2, b2, W3, b3, out, NPG);
}